// Circuit_87634512707722
// MI455X (gfx1250) — compile-verified
//
#include <hip/hip_runtime.h>
#include <hip/hip_bf16.h>
#include <math.h>

// v2f: A/B fragment of V_WMMA_F32_16X16X4_F32 (16x4 / 4x16 f32 -> 2 f32/lane)
// v8f: C/D fragment (16x16 f32 -> 8 f32/lane)
typedef __attribute__((ext_vector_type(2))) float v2f;
typedef __attribute__((ext_vector_type(8))) float v8f;

// ---------------------------------------------------------------------------
// Kernel 1: build the fused 16x16 real circuit matrix M' from theta.
//
// Complex 8x8 circuit matrix:
//   U[(a0,a1,a2),(b0,b1,b2)] = G0[a0][b0] * G1[a1][b1] * delta(b2, a2^a0)
// with Gq = RX(theta_q) * H = (1/sqrt2) [[alpha, beta],[alpha, -beta]],
//   alpha = c - i s, beta = c + i s,  c = cos(theta/2), s = sin(theta/2).
// Real 16x16 M' (rows interleaved re/im to match output layout (B,8,2)):
//   row 2i  : [ Re U[i][:] | -Im U[i][:] ]
//   row 2i+1: [ Im U[i][:] |  Re U[i][:] ]
// acting on v = [re0..re7, im0..im7].
// ---------------------------------------------------------------------------
__global__ void build_gate_matrix_kernel(const float* __restrict__ theta,
                                         float* __restrict__ M) {
  int t = threadIdx.x;                 // 256 threads -> one M' entry each
  int ri = t >> 4;                     // output row 0..15
  int cj = t & 15;                     // input col 0..15

  float c0 = cosf(theta[0] * 0.5f), s0 = sinf(theta[0] * 0.5f);
  float c1 = cosf(theta[1] * 0.5f), s1 = sinf(theta[1] * 0.5f);

  int i = ri >> 1, part = ri & 1;      // complex out index, 0=re / 1=im
  int j = cj & 7;                      // complex in index
  int a0 = (i >> 2) & 1, a1 = (i >> 1) & 1, a2 = i & 1;
  int b0 = (j >> 2) & 1, b1 = (j >> 1) & 1, b2 = j & 1;

  float val = 0.0f;
  if (b2 == (a2 ^ a0)) {               // CNOT(q0->q2) index permutation
    float g0r, g0i, g1r, g1i;
    if (b0 == 0) { g0r = c0; g0i = -s0; }                       // alpha0
    else { g0r = (a0 == 0) ? c0 : -c0; g0i = (a0 == 0) ? s0 : -s0; } // +-beta0
    if (b1 == 0) { g1r = c1; g1i = -s1; }                       // alpha1
    else { g1r = (a1 == 0) ? c1 : -c1; g1i = (a1 == 0) ? s1 : -s1; } // +-beta1
    // (1/sqrt2)^2 = 0.5 from the two Hadamard-fused gates
    float ur = 0.5f * (g0r * g1r - g0i * g1i);
    float ui = 0.5f * (g0r * g1i + g0i * g1r);
    bool imCol = (cj >= 8);
    val = (part == 0) ? (imCol ? -ui : ur) : (imCol ? ur : ui);
  }
  M[ri * 16 + cj] = val;
}

// ---------------------------------------------------------------------------
// Kernel 2: per 16-state tile, D(16x16) = M'(16x16) * V(16x16) via four
// V_WMMA_F32_16X16X4_F32 ops (K accumulated 4 at a time). Pure f32: exact FMAs.
//
// Fragment mapping (wave32):
//   A 16x4:  lane<16 -> M'[lane][4kc+0,1]; lane>=16 -> M'[lane-16][4kc+2,3]
//   B 4x16:  lane N = lane&15; half selects K pair -> lane reads v_n[4kc+kb+{0,1}]
//            kb = 2*(lane>>4); v_n = [x_real[n][0..7], x_imag[n][0..7]]
//   D 16x16: c[v] = out row (8*half + v) of state n -> two float4 stores
// Wave-uniform control flow keeps EXEC all-1s (required by WMMA).
// ---------------------------------------------------------------------------
__global__ void qsim_wmma_kernel(const float* __restrict__ xr,
                                 const float* __restrict__ xi,
                                 const float* __restrict__ M,
                                 float* __restrict__ out,
                                 int ntiles) {
  const int lane   = threadIdx.x & 31;
  const int wave   = (int)((blockIdx.x * blockDim.x + threadIdx.x) >> 5);
  const int nwaves = (int)((gridDim.x * blockDim.x) >> 5);
  const int half   = lane >> 4;        // 0 or 1
  const int mrow   = lane & 15;        // A row / B column (state) / D column
  const int kb     = half << 1;        // K base within each group of 4

  // A fragments (M' is tiny and shared -> hits L2/L0 for every wave)
  const float* mr = M + mrow * 16 + kb;
  v2f a0, a1, a2, a3;
  a0.x = mr[0];  a0.y = mr[1];         // kc=0: cols kb,   kb+1
  a1.x = mr[4];  a1.y = mr[5];         // kc=1: cols kb+4, kb+5
  a2.x = mr[8];  a2.y = mr[9];         // kc=2: cols kb+8, kb+9
  a3.x = mr[12]; a3.y = mr[13];        // kc=3: cols kb+12,kb+13

  for (int tile = wave; tile < ntiles; tile += nwaves) {
    const size_t n = (size_t)tile * 16 + (size_t)mrow;   // state for this lane
    const float* rp = xr + n * 8 + kb;
    const float* ip = xi + n * 8 + kb;
    // The wave's four 8B loads exactly tile 16 contiguous states' rows.
    v2f b0 = *(const v2f*)(rp);        // v[kb..kb+1]   = re[kb..kb+1]
    v2f b1 = *(const v2f*)(rp + 4);    // v[kb+4..]     = re[kb+4..]
    v2f b2 = *(const v2f*)(ip);        // v[kb+8..]     = im[kb..]
    v2f b3 = *(const v2f*)(ip + 4);    // v[kb+12..]    = im[kb+4..]

    v8f c = {};
    c = __builtin_amdgcn_wmma_f32_16x16x4_f32(false, a0, false, b0, (short)0, c, false, false);
    c = __builtin_amdgcn_wmma_f32_16x16x4_f32(false, a1, false, b1, (short)0, c, false, false);
    c = __builtin_amdgcn_wmma_f32_16x16x4_f32(false, a2, false, b2, (short)0, c, false, false);
    c = __builtin_amdgcn_wmma_f32_16x16x4_f32(false, a3, false, b3, (short)0, c, false, false);

    // Interleaved output rows: half 0 -> m=0..7, half 1 -> m=8..15
    float* op = out + n * 16 + (size_t)(half << 3);
    float4 lo; lo.x = c[0]; lo.y = c[1]; lo.z = c[2]; lo.w = c[3];
    float4 hi; hi.x = c[4]; hi.y = c[5]; hi.z = c[6]; hi.w = c[7];
    *(float4*)(op)     = lo;
    *(float4*)(op + 4) = hi;
  }
}

extern "C" void kernel_launch(void* const* d_in, const int* in_sizes, int n_in,
                              void* d_out, int out_size, void* d_ws, size_t ws_size,
                              hipStream_t stream) {
  const float* xr    = (const float*)d_in[0];   // x_real (B,8) f32
  const float* xi    = (const float*)d_in[1];   // x_imag (B,8) f32
  const float* theta = (const float*)d_in[2];   // (2,) f32
  // d_in[3] (angle) is dead in the reference.
  float* out = (float*)d_out;                   // (B,8,2) f32
  float* M   = (float*)d_ws;                    // 256 f32 = 1 KB scratch

  const int n_states = in_sizes[0] / 8;         // B = 2^21
  const int ntiles   = n_states / 16;           // 131072 (B divisible by 16)

  // Step 1: fuse the whole circuit into one 16x16 real matrix on-device.
  build_gate_matrix_kernel<<<1, 256, 0, stream>>>(theta, M);

  // Step 2: bandwidth-bound streaming GEMM; ~8 tiles per wave.
  const int threads = 256;                      // 8 waves per block
  int waves  = (ntiles + 7) / 8;
  int blocks = (waves * 32 + threads - 1) / threads;
  if (blocks < 1) blocks = 1;
  if (blocks > 65535) blocks = 65535;
  qsim_wmma_kernel<<<blocks, threads, 0, stream>>>(xr, xi, M, out, ntiles);
}